// InstantLoRAMidModule_2826088480798
// MI455X (gfx1250) — compile-verified
//
#include <hip/hip_runtime.h>

typedef __attribute__((ext_vector_type(2))) float v2f;
typedef __attribute__((ext_vector_type(8))) float v8f;

#define NB     8
#define NS     1024
#define DIN    1280
#define RANK   64
#define DOUT   1280
#define DOWN_SIZE (RANK * DIN)          // 81920
#define UP_SIZE   (DOUT * RANK)         // 81920
#define EMB_STRIDE (DOWN_SIZE + UP_SIZE)
#define HP     68                        // LDS row pitch (floats), bank-conflict padding

// Fused dynamic-LoRA: out[b,s,:] = (x[b,s,:] @ Wd^T) @ Wu^T, per-sample weights.
// One block = 4 waves handles one (b, 16-row) tile. Phase A computes the
// 16x64 h-tile with v_wmma_f32_16x16x4_f32 (wave w owns r-block w), stages it
// in LDS; phase B expands to 16x1280 output (wave w owns a 320-col slice).
__global__ __launch_bounds__(128) void lora_fused_kernel(
    const float* __restrict__ x,
    const float* __restrict__ embed,
    float* __restrict__ out) {
  __shared__ float h_lds[16 * HP];

  const int tid   = threadIdx.x;
  const int wave  = tid >> 5;      // 0..3
  const int lane  = tid & 31;
  const int lrow  = lane & 15;     // m (A) / n (B) within 16x16 tile
  const int lhalf = lane >> 4;     // K-half selector per ISA A/B layout

  const int tile = blockIdx.x;             // 0 .. B*S/16-1
  const int b    = tile >> 6;              // / (S/16 = 64)
  const int s0   = (tile & 63) << 4;

  const float* xb = x + ((size_t)b * NS + s0) * DIN;
  const float* wd = embed + (size_t)b * EMB_STRIDE;              // [RANK][DIN] row-major
  const float* wu = wd + DOWN_SIZE;                              // [DOUT][RANK] row-major

  // ---------------- Phase A: h tile (16 x 64), wave w -> r in [16w, 16w+16) ----
  // A frag: lane holds {x[s0+lrow][k + 2*lhalf], x[..][k + 2*lhalf + 1]}
  // B frag: lane holds {wd[16w+lrow][k + 2*lhalf], wd[..][k + 2*lhalf + 1]}
  const float* aPtr = xb + (size_t)lrow * DIN + 2 * lhalf;
  const float* bPtr = wd + (size_t)(wave * 16 + lrow) * DIN + 2 * lhalf;

  v8f acc0 = {};
  v8f acc1 = {};
  for (int k = 0; k < DIN; k += 8) {     // 160 iters, 2 independent WMMA chains
    v2f a0 = *(const v2f*)(aPtr + k);
    v2f b0 = *(const v2f*)(bPtr + k);
    v2f a1 = *(const v2f*)(aPtr + k + 4);
    v2f b1 = *(const v2f*)(bPtr + k + 4);
    acc0 = __builtin_amdgcn_wmma_f32_16x16x4_f32(false, a0, false, b0,
                                                 (short)0, acc0, false, false);
    acc1 = __builtin_amdgcn_wmma_f32_16x16x4_f32(false, a1, false, b1,
                                                 (short)0, acc1, false, false);
  }
  v8f hacc = acc0 + acc1;

  // C/D layout: VGPR v -> row (v + 8*lhalf), col lrow. Stage h in LDS [16][HP].
#pragma unroll
  for (int v = 0; v < 8; ++v) {
    h_lds[(v + 8 * lhalf) * HP + wave * 16 + lrow] = hacc[v];
  }
  __syncthreads();

  // ---------------- Phase B: out tile 16 x 1280, wave w -> cols [320w, 320w+320) ----
  // Preload all A frags of h (K = 64 -> 16 wmma steps) from LDS once.
  v2f ha[16];
#pragma unroll
  for (int t = 0; t < 16; ++t) {
    int k = t * 4 + 2 * lhalf;
    ha[t] = *(const v2f*)&h_lds[lrow * HP + k];
  }

  float* outRow = out + ((size_t)b * NS + s0) * DOUT;
  for (int nt = 0; nt < 20; ++nt) {
    const int n0 = wave * 320 + nt * 16;
    const float* wPtr = wu + (size_t)(n0 + lrow) * RANK + 2 * lhalf;
    v8f c0 = {};
    v8f c1 = {};
#pragma unroll
    for (int t = 0; t < 16; t += 2) {
      v2f b0 = *(const v2f*)(wPtr + t * 4);
      v2f b1 = *(const v2f*)(wPtr + t * 4 + 4);
      c0 = __builtin_amdgcn_wmma_f32_16x16x4_f32(false, ha[t], false, b0,
                                                 (short)0, c0, false, false);
      c1 = __builtin_amdgcn_wmma_f32_16x16x4_f32(false, ha[t + 1], false, b1,
                                                 (short)0, c1, false, false);
    }
    v8f c = c0 + c1;
#pragma unroll
    for (int v = 0; v < 8; ++v) {
      outRow[(size_t)(v + 8 * lhalf) * DOUT + n0 + lrow] = c[v];
    }
  }
}

extern "C" void kernel_launch(void* const* d_in, const int* in_sizes, int n_in,
                              void* d_out, int out_size, void* d_ws, size_t ws_size,
                              hipStream_t stream) {
  (void)in_sizes; (void)n_in; (void)out_size; (void)d_ws; (void)ws_size;
  const float* x     = (const float*)d_in[0];
  const float* embed = (const float*)d_in[1];
  float* out         = (float*)d_out;

  dim3 grid(NB * (NS / 16));   // 512 blocks
  dim3 block(128);             // 4 waves (wave32)
  hipLaunchKernelGGL(lora_fused_kernel, grid, block, 0, stream, x, embed, out);
}